// HeteroTextGCN_55800215109809
// MI455X (gfx1250) — compile-verified
//
#include <hip/hip_runtime.h>
#include <hip/hip_bf16.h>

typedef __attribute__((ext_vector_type(2))) float v2f;
typedef __attribute__((ext_vector_type(8))) float v8f;

// ---------------- utility kernels ----------------

__global__ void zero_kernel(float* __restrict__ p, int n) {
    int i = blockIdx.x * blockDim.x + threadIdx.x;
    if (i < n) p[i] = 0.0f;
}

// Count out-degree (src) and in-degree (dst) for all relations at once.
// src/dst are flat [nRel*E]; deg buffers are [nRel*N].
__global__ void degree_kernel(const int* __restrict__ src, const int* __restrict__ dst,
                              float* __restrict__ degO, float* __restrict__ degI,
                              int total, int N, int nE) {
    int i = blockIdx.x * blockDim.x + threadIdx.x;
    if (i < total) {
        int r = i / nE;
        atomicAdd(&degO[(size_t)r * N + src[i]], 1.0f);
        atomicAdd(&degI[(size_t)r * N + dst[i]], 1.0f);
    }
}

// In-place: v -> rsqrt(max(v, 1))
__global__ void rsqrt_kernel(float* __restrict__ p, int n) {
    int i = blockIdx.x * blockDim.x + threadIdx.x;
    if (i < n) p[i] = rsqrtf(fmaxf(p[i], 1.0f));
}

// out[n][c] = b[0][c] + b[1][c] + b[2][c]  (sum of per-relation biases)
__global__ void init_bias3_kernel(float* __restrict__ out, const float* __restrict__ b,
                                  int total, int C) {
    int i = blockIdx.x * blockDim.x + threadIdx.x;
    if (i < total) {
        int c = i % C;
        out[i] = b[c] + b[C + c] + b[2 * C + c];
    }
}

__global__ void leaky_kernel(float* __restrict__ p, int n) {
    int i = blockIdx.x * blockDim.x + threadIdx.x;
    if (i < n) {
        float v = p[i];
        p[i] = v >= 0.0f ? v : 0.01f * v;
    }
}

// ---------------- WMMA GEMM: H = (X .* scale[row]) @ W ----------------
// X: [Nrows, K] f32, W: [K, 128] f32, H: [Nrows, 128] f32.
// Block = 256 threads (8 wave32). Block tile: 128 rows x 128 cols.
// Wave w computes rows [m0+16w, m0+16w+16) x all 128 cols (8 WMMA col tiles).
//
// W chunk is stored k-pair interleaved in LDS: element (k, col) lives at
//   Ws2[(k>>1)*WS2 + 2*col + (k&1)]
// so a B fragment {W[k][col], W[k+1][col]} is ONE aligned ds_load_b64 into an
// even VGPR pair (no v_mov fix-ups before v_wmma). WS2 = 288 (mod 64 == 32)
// makes lanes 0-15 (pair-row p) and 16-31 (pair-row p+1) hit disjoint bank
// halves -> conflict-free.
template <int K>
__global__ __launch_bounds__(256) void gemm_wmma_kernel(
    const float* __restrict__ X, const float* __restrict__ scale,
    const float* __restrict__ W, float* __restrict__ H, int Nrows) {
    constexpr int TILE_M = 128;
    constexpr int TILE_K = 32;
    constexpr int XS = 36;    // padded stride for Xs (floats): conflict-free A-frag b64 reads
    constexpr int WS2 = 288;  // padded stride for Ws2 (floats) per k-pair row
    __shared__ float Xs[TILE_M * XS];            // 18432 B
    __shared__ float Ws2[(TILE_K / 2) * WS2];    // 18432 B

    const int tid = threadIdx.x;
    const int wave = tid >> 5;
    const int lane = tid & 31;
    const int lrow = lane & 15;   // row/col within 16-wide fragment
    const int khalf = lane >> 4;  // 0: K{0,1}, 1: K{2,3}
    const int m0 = blockIdx.x * TILE_M;

    v8f acc[8];
#pragma unroll
    for (int t = 0; t < 8; ++t) {
        v8f z = {};
        acc[t] = z;
    }

    for (int kc = 0; kc < K; kc += TILE_K) {
        __syncthreads();
        // Stage X chunk (scaled): 128 rows x 32 k = 4096 floats, float4 per thread x4
#pragma unroll
        for (int it = 0; it < (TILE_M * TILE_K) / (256 * 4); ++it) {
            int idx = (it * 256 + tid) * 4;
            int row = idx / TILE_K;
            int col = idx % TILE_K;
            int grow = m0 + row;
            float4 v = make_float4(0.0f, 0.0f, 0.0f, 0.0f);
            if (grow < Nrows) {
                float4 s4 = *reinterpret_cast<const float4*>(X + (size_t)grow * K + kc + col);
                float sc = scale[grow];
                v = make_float4(s4.x * sc, s4.y * sc, s4.z * sc, s4.w * sc);
            }
            *reinterpret_cast<float4*>(&Xs[row * XS + col]) = v;
        }
        // Stage W chunk k-pair interleaved: 16 pair-rows x 64 col-pairs = 1024 units,
        // each unit = 4 elements handled by one thread (2 global b64 loads, 2 LDS b64 stores).
#pragma unroll
        for (int it = 0; it < (TILE_K / 2 * 64) / 256; ++it) {
            int u = it * 256 + tid;
            int p = u >> 6;              // pair-row 0..15
            int c = (u & 63) * 2;        // col 0..126 step 2
            v2f g0 = *reinterpret_cast<const v2f*>(W + (size_t)(kc + 2 * p + 0) * 128 + c);
            v2f g1 = *reinterpret_cast<const v2f*>(W + (size_t)(kc + 2 * p + 1) * 128 + c);
            v2f s0;  s0.x = g0.x;  s0.y = g1.x;   // (k=2p, c), (k=2p+1, c)
            v2f s1;  s1.x = g0.y;  s1.y = g1.y;   // (k=2p, c+1), (k=2p+1, c+1)
            *reinterpret_cast<v2f*>(&Ws2[p * WS2 + 2 * c]) = s0;
            *reinterpret_cast<v2f*>(&Ws2[p * WS2 + 2 * c + 2]) = s1;
        }
        __syncthreads();

#pragma unroll
        for (int ks = 0; ks < TILE_K; ks += 4) {
            // A fragment (16x4 f32): lane<16 holds K{ks,ks+1}, lane>=16 holds K{ks+2,ks+3}
            const int arow = wave * 16 + lrow;
            v2f a = *reinterpret_cast<const v2f*>(&Xs[arow * XS + ks + 2 * khalf]);
            const int bp = (ks >> 1) + khalf;  // pair-row holding K{ks+2*khalf, ks+2*khalf+1}
#pragma unroll
            for (int t = 0; t < 8; ++t) {
                const int col = t * 16 + lrow;
                v2f b = *reinterpret_cast<const v2f*>(&Ws2[bp * WS2 + 2 * col]);
                acc[t] = __builtin_amdgcn_wmma_f32_16x16x4_f32(
                    false, a, false, b, (short)0, acc[t], false, false);
            }
        }
    }

    // Epilogue: C/D layout -> row = m0 + 16*wave + i + 8*khalf, col = 16*t + lrow
    const int rbase = m0 + wave * 16 + 8 * khalf;
#pragma unroll
    for (int t = 0; t < 8; ++t) {
#pragma unroll
        for (int i = 0; i < 8; ++i) {
            int row = rbase + i;
            if (row < Nrows) H[(size_t)row * 128 + t * 16 + lrow] = acc[t][i];
        }
    }
}

// ---------------- edge scatter: acc[dst] += H[src] * invI[dst] ----------------
// One edge per wave32; each lane handles 4 consecutive floats (128 cols total).
__global__ void scatter_kernel(const float* __restrict__ H, const int* __restrict__ src,
                               const int* __restrict__ dst, const float* __restrict__ invI,
                               float* __restrict__ acc, int nE) {
    int gid = blockIdx.x * blockDim.x + threadIdx.x;
    int e = gid >> 5;
    int lane = gid & 31;
    if (e < nE) {
        int s = src[e];
        int d = dst[e];
        float w = invI[d];
        float4 hv = *reinterpret_cast<const float4*>(H + (size_t)s * 128 + lane * 4);
        float* out = acc + (size_t)d * 128 + lane * 4;
        atomicAdd(out + 0, hv.x * w);
        atomicAdd(out + 1, hv.y * w);
        atomicAdd(out + 2, hv.z * w);
        atomicAdd(out + 3, hv.w * w);
    }
}

// ---------------- FC head: logits = h @ fcW + fcb ----------------
__global__ void fc_kernel(const float* __restrict__ h, const float* __restrict__ W,
                          const float* __restrict__ b, float* __restrict__ out, int N) {
    int i = blockIdx.x * blockDim.x + threadIdx.x;
    if (i < N * 16) {
        int n = i >> 4;
        int j = i & 15;
        float s = b[j];
        const float* hr = h + (size_t)n * 128;
#pragma unroll 16
        for (int k = 0; k < 128; ++k) s += hr[k] * W[k * 16 + j];
        out[i] = s;
    }
}

// ---------------- launch ----------------
extern "C" void kernel_launch(void* const* d_in, const int* in_sizes, int n_in,
                              void* d_out, int out_size, void* d_ws, size_t ws_size,
                              hipStream_t stream) {
    const float* x   = (const float*)d_in[0];
    const int*   src = (const int*)d_in[1];
    const int*   dst = (const int*)d_in[2];
    const float* W0  = (const float*)d_in[3];
    const float* b0  = (const float*)d_in[4];
    const float* W1  = (const float*)d_in[5];
    const float* b1  = (const float*)d_in[6];
    const float* fcW = (const float*)d_in[7];
    const float* fcb = (const float*)d_in[8];

    const int nRel = 3;
    const int N = in_sizes[0] / 256;    // 100000 nodes
    const int E = in_sizes[1] / nRel;   // 600000 edges per relation
    const int dH = 128;

    // Workspace layout (~105 MB)
    float* invO = (float*)d_ws;                    // [nRel*N] out-degree^-1/2
    float* invI = invO + (size_t)nRel * N;         // [nRel*N] in-degree^-1/2
    float* Hbuf = invI + (size_t)nRel * N;         // [N,128] per-relation transformed feats
    float* L1   = Hbuf + (size_t)N * dH;           // [N,128] layer-1 accumulator
    float* outH   = (float*)d_out;                 // [N,128] final h (output 0)
    float* logits = outH + (size_t)N * dH;         // [N,16]  logits (output 1)

    const int nDeg = 2 * nRel * N;  // invO+invI are contiguous
    zero_kernel<<<(nDeg + 255) / 256, 256, 0, stream>>>(invO, nDeg);

    const int totE = nRel * E;
    degree_kernel<<<(totE + 255) / 256, 256, 0, stream>>>(src, dst, invO, invI, totE, N, E);
    rsqrt_kernel<<<(nDeg + 255) / 256, 256, 0, stream>>>(invO, nDeg);

    const int nNC = N * dH;
    const int gemmBlocks = (N + 127) / 128;
    const int scatBlocks = (E * 32 + 255) / 256;

    // ---- layer 1: x[N,256] -> L1[N,128] ----
    init_bias3_kernel<<<(nNC + 255) / 256, 256, 0, stream>>>(L1, b0, nNC, dH);
    for (int r = 0; r < nRel; ++r) {
        gemm_wmma_kernel<256><<<gemmBlocks, 256, 0, stream>>>(
            x, invO + (size_t)r * N, W0 + (size_t)r * 256 * 128, Hbuf, N);
        scatter_kernel<<<scatBlocks, 256, 0, stream>>>(
            Hbuf, src + (size_t)r * E, dst + (size_t)r * E, invI + (size_t)r * N, L1, E);
    }
    leaky_kernel<<<(nNC + 255) / 256, 256, 0, stream>>>(L1, nNC);

    // ---- layer 2: L1[N,128] -> outH[N,128] (no activation) ----
    init_bias3_kernel<<<(nNC + 255) / 256, 256, 0, stream>>>(outH, b1, nNC, dH);
    for (int r = 0; r < nRel; ++r) {
        gemm_wmma_kernel<128><<<gemmBlocks, 256, 0, stream>>>(
            L1, invO + (size_t)r * N, W1 + (size_t)r * 128 * 128, Hbuf, N);
        scatter_kernel<<<scatBlocks, 256, 0, stream>>>(
            Hbuf, src + (size_t)r * E, dst + (size_t)r * E, invI + (size_t)r * N, outH, E);
    }

    // ---- FC head: logits[N,16] ----
    const int nL = N * 16;
    fc_kernel<<<(nL + 255) / 256, 256, 0, stream>>>(outH, fcW, fcb, logits, N);
}